// CrossModalCenterLoss_31353261261657
// MI455X (gfx1250) — compile-verified
//
#include <hip/hip_runtime.h>
#include <hip/hip_bf16.h>

#define BATCH       2048
#define NUM_CLASSES 50000
#define FEAT_DIM    256
#define CLAMP_MIN   1e-12f
#define CLAMP_MAX   1e12f

typedef __attribute__((ext_vector_type(2))) float v2f;
typedef __attribute__((ext_vector_type(8))) float v8f;

// Each wave32 handles 16 batch rows. For the symmetric Gram computation
// D = diff * diff^T with the 16x16x4 f32 WMMA, the A-matrix (16x4) and
// B-matrix (4x16) VGPR layouts coincide: lane L (L<16) holds k=0,1 of row
// L%16 in its two VGPRs, lane L+16 holds k=2,3. So one float2 per lane per
// k-chunk feeds both operands. Accumulate 64 chunks (256/4); diagonal of the
// 16x16 f32 accumulator is the squared distance per row.
__global__ void __launch_bounds__(256)
center_dist_kernel(const float* __restrict__ x,
                   const int* __restrict__ labels,
                   const float* __restrict__ centers,
                   float* __restrict__ row_d) {
    const int lane    = threadIdx.x & 31;
    const int wave    = threadIdx.x >> 5;
    const int rowBase = (blockIdx.x * 8 + wave) * 16;
    const int m       = lane & 15;   // which of the 16 rows this lane serves
    const int khalf   = lane >> 4;   // 0 -> k={0,1}, 1 -> k={2,3} within chunk

    const int row = rowBase + m;
    const int lab = labels[row];
    const float* xr = x       + (size_t)row * FEAT_DIM + khalf * 2;
    const float* cr = centers + (size_t)lab * FEAT_DIM + khalf * 2;

    v8f acc = {};
#pragma unroll
    for (int kc = 0; kc < FEAT_DIM / 4; ++kc) {
        v2f xv = *(const v2f*)(xr + kc * 4);
        v2f cv = *(const v2f*)(cr + kc * 4);
        v2f dv = xv - cv;
        // D = A*B + C with A == B == diff tile (symmetric Gram).
        acc = __builtin_amdgcn_wmma_f32_16x16x4_f32(
            /*neg_a=*/false, dv, /*neg_b=*/false, dv,
            /*c_mod=*/(short)0, acc, /*reuse_a=*/false, /*reuse_b=*/false);
    }

    // Diagonal extraction per the 16x16 f32 C/D layout:
    //   m < 8 : VGPR m,   lane m        -> lanes 0..7 read acc[lane]
    //   m >= 8: VGPR m-8, lane m+16     -> lanes 24..31 read acc[lane-24]
    const bool active = (lane < 8) || (lane >= 24);
    const int  idx    = (lane < 8) ? lane : (lane - 24);
    float sel = acc[0];
    sel = (idx == 1) ? acc[1] : sel;
    sel = (idx == 2) ? acc[2] : sel;
    sel = (idx == 3) ? acc[3] : sel;
    sel = (idx == 4) ? acc[4] : sel;
    sel = (idx == 5) ? acc[5] : sel;
    sel = (idx == 6) ? acc[6] : sel;
    sel = (idx == 7) ? acc[7] : sel;

    if (active) {
        const int mOut = (lane < 8) ? lane : (lane - 16);
        float dclip = fminf(fmaxf(sel, CLAMP_MIN), CLAMP_MAX);
        row_d[rowBase + mOut] = dclip;
    }
}

// Deterministic single-block tree reduction of the 2048 per-row distances.
// All (B*C - B) masked-off entries clamp to exactly 1e-12, contributing the
// closed-form constant (C-1)*1e-12 after the /B normalization.
__global__ void __launch_bounds__(256)
center_loss_reduce_kernel(const float* __restrict__ row_d,
                          float* __restrict__ out) {
    __shared__ float s[256];
    float sum = 0.0f;
#pragma unroll
    for (int i = 0; i < BATCH / 256; ++i)
        sum += row_d[threadIdx.x + i * 256];
    s[threadIdx.x] = sum;
    __syncthreads();
#pragma unroll
    for (int off = 128; off > 0; off >>= 1) {
        if ((int)threadIdx.x < off) s[threadIdx.x] += s[threadIdx.x + off];
        __syncthreads();
    }
    if (threadIdx.x == 0) {
        out[0] = s[0] / (float)BATCH
               + (float)(NUM_CLASSES - 1) * CLAMP_MIN;
    }
}

extern "C" void kernel_launch(void* const* d_in, const int* in_sizes, int n_in,
                              void* d_out, int out_size, void* d_ws, size_t ws_size,
                              hipStream_t stream) {
    const float* x       = (const float*)d_in[0];
    const int*   labels  = (const int*)d_in[1];
    const float* centers = (const float*)d_in[2];
    float*       out     = (float*)d_out;
    float*       row_d   = (float*)d_ws;   // BATCH floats of scratch

    (void)in_sizes; (void)n_in; (void)out_size; (void)ws_size;

    // 16 blocks x 8 waves x 16 rows = 2048 rows
    center_dist_kernel<<<16, 256, 0, stream>>>(x, labels, centers, row_d);
    center_loss_reduce_kernel<<<1, 256, 0, stream>>>(row_d, out);
}